// GaussianTensor_59442347377427
// MI455X (gfx1250) — compile-verified
//
#include <hip/hip_runtime.h>
#include <hip/hip_bf16.h>

typedef __attribute__((ext_vector_type(2))) float v2f;
typedef __attribute__((ext_vector_type(4))) float v4f;
typedef __attribute__((ext_vector_type(8))) float v8f;

#define GSIZE 64  // feature / mixture dimension

// -0.5 * 64 * log(2*pi)
#define NEG_HALF_D_LOG2PI (-58.8120661250990542f)

// One wave (32 lanes) handles one 16-row tile of one batch:
//   rows j = t*16 .. t*16+15, full k = 0..63.
// A-matrix (16x4 f32) lane layout: lanes 0-15 hold row=lane, K={0,1} in v0,v1;
// lanes 16-31 hold row=lane-16, K={2,3}. So each lane loads a contiguous
// float2 of mu per 4-wide k-chunk, forms d^2, and a ones-B WMMA accumulates
// the row sums: D[j][n] += sum_k A[j][k].
__global__ __launch_bounds__(256) void gauss_logpdf_wmma(
    const float* __restrict__ inputs,  // [B, 64]
    const float* __restrict__ means,   // [B, 64*64] row-major mu[b][j][k]
    float* __restrict__ out,           // [B, 64]
    int nBatches)
{
    const int lane        = threadIdx.x & 31;
    const int waveInBlock = threadIdx.x >> 5;
    const int tileId      = blockIdx.x * 8 + waveInBlock; // 4 tiles per batch
    const int b           = tileId >> 2;
    const int t           = tileId & 3;
    if (b >= nBatches) return;  // wave-uniform guard (never taken for exact grid)

    const int row  = lane & 15;          // j within tile
    const int kOff = (lane >> 4) << 1;   // 0 for lanes 0-15, 2 for lanes 16-31

    const float* mu = means + (size_t)b * (GSIZE * GSIZE)
                            + (size_t)(t * 16 + row) * GSIZE + kOff;
    const float* xp = inputs + (size_t)b * GSIZE + kOff;

    v2f bOnes;
    bOnes.x = 1.0f;
    bOnes.y = 1.0f;

    v8f acc = {};  // 16x16 f32 accumulator: row sums replicated across columns

#pragma unroll
    for (int kc = 0; kc < 16; ++kc) {  // k chunk base = kc*4
        const v2f m  = *(const v2f*)(mu + kc * 4);
        const v2f xv = *(const v2f*)(xp + kc * 4);
        v2f a;
        a.x = xv.x - m.x;
        a.y = xv.y - m.y;
        a.x = a.x * a.x;
        a.y = a.y * a.y;
        // D = A(16x4) * ones(4x16) + D  => D[j][n] += sum of 4 k-values of row j
        acc = __builtin_amdgcn_wmma_f32_16x16x4_f32(
            /*neg_a=*/false, a, /*neg_b=*/false, bOnes,
            /*c_mod=*/(short)0, acc, /*reuse_a=*/false, /*reuse_b=*/false);
    }

    // C/D layout: VGPR r holds M=r (lanes 0-15, N=lane) and M=r+8 (lanes 16-31).
    // Column N=0 lives in lane 0 (M=0..7) and lane 16 (M=8..15); every column
    // holds the same row-sum, so lanes 0 and 16 each store 8 results.
    if (row == 0) {
        float* o = out + (size_t)b * GSIZE + t * 16 + (lane >> 4) * 8;
        v4f o0, o1;
        o0.x = fmaf(-0.5f, acc[0], NEG_HALF_D_LOG2PI);
        o0.y = fmaf(-0.5f, acc[1], NEG_HALF_D_LOG2PI);
        o0.z = fmaf(-0.5f, acc[2], NEG_HALF_D_LOG2PI);
        o0.w = fmaf(-0.5f, acc[3], NEG_HALF_D_LOG2PI);
        o1.x = fmaf(-0.5f, acc[4], NEG_HALF_D_LOG2PI);
        o1.y = fmaf(-0.5f, acc[5], NEG_HALF_D_LOG2PI);
        o1.z = fmaf(-0.5f, acc[6], NEG_HALF_D_LOG2PI);
        o1.w = fmaf(-0.5f, acc[7], NEG_HALF_D_LOG2PI);
        *(v4f*)(o)     = o0;
        *(v4f*)(o + 4) = o1;
    }
}

extern "C" void kernel_launch(void* const* d_in, const int* in_sizes, int n_in,
                              void* d_out, int out_size, void* d_ws, size_t ws_size,
                              hipStream_t stream) {
    const float* inputs = (const float*)d_in[0];  // (32768, 64) f32
    const float* means  = (const float*)d_in[1];  // (32768, 4096) f32
    float* out = (float*)d_out;                   // (32768, 64) f32

    const int nBatches = in_sizes[0] / GSIZE;     // 32768
    const int nTiles   = nBatches * 4;            // one wave per 16-row tile
    const int nBlocks  = (nTiles + 7) / 8;        // 8 waves (256 threads) per block

    gauss_logpdf_wmma<<<nBlocks, 256, 0, stream>>>(inputs, means, out, nBatches);
}